// Quantizer_78658031059423
// MI455X (gfx1250) — compile-verified
//
#include <hip/hip_runtime.h>
#include <hip/hip_bf16.h>

typedef __attribute__((ext_vector_type(2))) float        v2f;
typedef __attribute__((ext_vector_type(4))) float        v4f;
typedef __attribute__((ext_vector_type(8))) float        v8f;
typedef __attribute__((ext_vector_type(4))) unsigned int v4u;
typedef __attribute__((ext_vector_type(4))) int          v4i;
typedef __attribute__((ext_vector_type(8))) int          v8i;

#define NROWS   65536
#define DIM     256
#define NCODE   2048
#define NTILES  (NCODE / 16)

// ---------------------------------------------------------------------------
// Phase 0: seed[k] = -0.5 * ||e_k||^2   (argmin dist == argmax (x.e + seed))
// ---------------------------------------------------------------------------
__global__ void vq_seed_kernel(const float* __restrict__ cb, float* __restrict__ seed) {
    int k = blockIdx.x * 256 + threadIdx.x;   // 2048 threads, coalesced over k
    float s = 0.f;
    for (int d = 0; d < DIM; ++d) {
        float v = cb[(long)d * NCODE + k];
        s += v * v;
    }
    seed[k] = -0.5f * s;
}

__global__ void vq_init_kernel(int* __restrict__ counts) {
    int i = blockIdx.x * 256 + threadIdx.x;
    if (i < NCODE) counts[i] = 0;
}

// ---------------------------------------------------------------------------
// TDM descriptor issue: copy one 16-code x 256-D codebook tile (memory order,
// 16 KB contiguous) into LDS.  2-D D#: tile_dim0=16 (contiguous codes),
// tile_dim1=256 (D rows), tensor_dim0_stride=2048.  Groups 2/3 unused (zero).
// ---------------------------------------------------------------------------
__device__ __forceinline__ void tdm_load_tile(const float* gsrc, float* lbuf) {
    unsigned long ga  = (unsigned long)gsrc;
    unsigned int  lds = (unsigned int)(unsigned long)lbuf;  // low 32 bits = LDS offset

    v4u g0;
    g0.x = 1u;                                   // count=1, user descriptor
    g0.y = lds;                                  // lds_addr
    g0.z = (unsigned int)ga;                     // global_addr[31:0]
    g0.w = (unsigned int)((ga >> 32) & 0x01FFFFFFu) | 0x80000000u;  // addr[56:32] | type=2

    v8i g1;
    g1[0] = (int)(2u << 16);          // workgroup_mask=0, data_size=2 (4 bytes)
    g1[1] = (int)(2048u << 16);       // tensor_dim0[15:0]=2048 (bits 63:48)
    g1[2] = (int)(256u  << 16);       // tensor_dim0 hi=0, tensor_dim1[15:0]=256
    g1[3] = (int)(16u   << 16);       // tensor_dim1 hi=0, tile_dim0=16
    g1[4] = 256;                      // tile_dim1=256, tile_dim2=0
    g1[5] = 2048;                     // tensor_dim0_stride lo32
    g1[6] = 0;                        // stride0 hi, stride1 lo
    g1[7] = 0;                        // stride1 hi

    v4i z4 = {0, 0, 0, 0};
    v8i z8 = {0, 0, 0, 0, 0, 0, 0, 0};
    __builtin_amdgcn_tensor_load_to_lds(g0, g1, z4, z4, z8, 0);
}

// ---------------------------------------------------------------------------
// Phase 1: nearest-code search via V_WMMA_F32_16X16X4_F32 + TDM double-buffer.
// Block = 8 waves, 128 rows. Wave w owns rows [base+16w, base+16w+16).
// A = 16x4 codebook tile (codes as M; TDM-staged LDS, shared by all waves)
// B = 4x16 x-rows       (held in 128 VGPRs for the entire K=2048 sweep)
// C = 16x16 scores, seeded with -0.5*||e||^2; running argmax per lane.
// ---------------------------------------------------------------------------
__global__ __launch_bounds__(256, 1)
void vq_argmax_kernel(const float* __restrict__ x,
                      const float* __restrict__ cb,
                      const float* __restrict__ seed,
                      int* __restrict__ idx_out) {
    __shared__ float buf[2][DIM * 16];   // 2 x 16 KB, layout [d][code] (memory order)

    const int t    = threadIdx.x;
    const int lane = t & 31;
    const int wave = t >> 5;
    const int n16  = lane & 15;   // row-in-tile (N) and code-in-tile (M)
    const int hi   = lane >> 4;   // lane half selects K pair / M half
    const long row = (long)blockIdx.x * 128 + wave * 16 + n16;

    // Prefetch tile 0 via TDM (wave 0 only; TENSORcnt is per-wave).
    if (wave == 0) tdm_load_tile(cb, &buf[0][0]);

    // B operand registers: xb[s] = x[row][4s+2hi .. 4s+2hi+1]   (64 x v2f = 128 VGPRs)
    v2f xb[64];
    {
        const float* xr = x + row * DIM + 2 * hi;
        #pragma unroll
        for (int s = 0; s < 64; ++s)
            xb[s] = *(const v2f*)(xr + 4 * s);
    }

    float bestVal = -3.4e38f;
    int   bestIdx = 0;

    for (int tl = 0; tl < NTILES; ++tl) {
        if (wave == 0)
            __builtin_amdgcn_s_wait_tensorcnt(0);   // tile tl landed in LDS
        __syncthreads();                            // visible to all; buf[tl+1 & 1] free
        if (wave == 0 && tl + 1 < NTILES)
            tdm_load_tile(cb + (tl + 1) * 16, &buf[(tl + 1) & 1][0]);

        const float* B = &buf[tl & 1][0];

        // Seed accumulator with -0.5*||e||^2 for this lane's 8 codes.
        v8f c;
        #pragma unroll
        for (int j = 0; j < 8; ++j)
            c[j] = seed[tl * 16 + j + 8 * hi];

        // 64 WMMA steps over K=256 (4 per step).
        // A element (m=n16, d): B[d*16 + n16]; lane half picks d = 4s+2hi(+1).
        const float* aPtr = B + 2 * hi * 16 + n16;
        #pragma unroll
        for (int s = 0; s < 64; ++s) {
            v2f a;
            a.x = aPtr[s * 64];        // d = 4s + 2hi
            a.y = aPtr[s * 64 + 16];   // d = 4s + 2hi + 1
            c = __builtin_amdgcn_wmma_f32_16x16x4_f32(
                    /*neg_a=*/false, a, /*neg_b=*/false, xb[s],
                    /*c_mod=*/(short)0, c, /*reuse_a=*/false, /*reuse_b=*/false);
        }

        // Running argmax (codes visited in ascending order -> ties keep first).
        #pragma unroll
        for (int j = 0; j < 8; ++j) {
            int code = tl * 16 + j + 8 * hi;
            if (c[j] > bestVal) { bestVal = c[j]; bestIdx = code; }
        }
    }

    // Merge the two M-halves (lane l holds codes {..+8hi}; partner l^16 the rest).
    float oVal = __shfl_xor(bestVal, 16, 32);
    int   oIdx = __shfl_xor(bestIdx, 16, 32);
    if (oVal > bestVal || (oVal == bestVal && oIdx < bestIdx)) {
        bestVal = oVal; bestIdx = oIdx;
    }
    if (hi == 0) idx_out[row] = bestIdx;
}

// ---------------------------------------------------------------------------
// Phase 2: gather quantized, straight-through output, per-block loss partial,
// code histogram. 8 rows/block, 32 lanes/row, 8 d per lane.
// ---------------------------------------------------------------------------
__global__ __launch_bounds__(256)
void vq_quantize_kernel(const float* __restrict__ x,
                        const float* __restrict__ cb,
                        const int* __restrict__ idx,
                        float* __restrict__ out,
                        int* __restrict__ counts,
                        float* __restrict__ partial) {
    __shared__ float wsum[8];
    const int t    = threadIdx.x;
    const int lane = t & 31;
    const int wave = t >> 5;                       // row within block
    const long row = (long)blockIdx.x * 8 + wave;
    const int code = idx[row];

    const float* xr   = x + row * DIM;
    float*       outr = out + row * DIM;

    float local = 0.f;
    #pragma unroll
    for (int i = 0; i < 8; ++i) {
        int d = i * 32 + lane;                     // coalesced x/out access
        float q  = cb[(long)d * NCODE + code];     // gather (codebook is L2-resident)
        float xv = xr[d];
        float diff = xv - q;
        local += diff * diff;
        outr[d] = xv + (q - xv);                   // straight-through estimator
    }
    // wave reduction
    #pragma unroll
    for (int off = 16; off; off >>= 1)
        local += __shfl_xor(local, off, 32);
    if (lane == 0) {
        wsum[wave] = local;
        atomicAdd(&counts[code], 1);
    }
    __syncthreads();
    if (t == 0) {
        float s = 0.f;
        #pragma unroll
        for (int w = 0; w < 8; ++w) s += wsum[w];
        partial[blockIdx.x] = s;                   // deterministic partial sums
    }
}

// ---------------------------------------------------------------------------
// Phase 3: finalize loss (mean of 1.25*(x-q)^2) and perplexity.
// ---------------------------------------------------------------------------
__global__ void vq_finalize_kernel(const int* __restrict__ counts,
                                   const float* __restrict__ partial,
                                   float* __restrict__ out_loss,
                                   float* __restrict__ out_perp) {
    __shared__ float red[256];
    const int t = threadIdx.x;

    float ls = 0.f;
    for (int i = t; i < 8192; i += 256) ls += partial[i];
    red[t] = ls; __syncthreads();
    for (int off = 128; off; off >>= 1) {
        if (t < off) red[t] += red[t + off];
        __syncthreads();
    }
    float lossTotal = red[0];
    __syncthreads();

    float es = 0.f;
    for (int i = t; i < NCODE; i += 256) {
        float p = (float)counts[i] * (1.0f / 65536.0f);
        es += p * logf(p + 1e-10f);
    }
    red[t] = es; __syncthreads();
    for (int off = 128; off; off >>= 1) {
        if (t < off) red[t] += red[t + off];
        __syncthreads();
    }
    if (t == 0) {
        *out_loss = lossTotal * (1.25f / 16777216.0f);
        *out_perp = expf(-red[0]);
    }
}

// ---------------------------------------------------------------------------
extern "C" void kernel_launch(void* const* d_in, const int* in_sizes, int n_in,
                              void* d_out, int out_size, void* d_ws, size_t ws_size,
                              hipStream_t stream) {
    const float* x  = (const float*)d_in[0];   // (64,32,32,256) = 65536 x 256
    const float* cb = (const float*)d_in[1];   // (256, 2048)
    float* out = (float*)d_out;                // [quantized_st | loss | perplexity]

    char* ws = (char*)d_ws;
    float* seed    = (float*)(ws);                           //  8 KB
    int*   idx     = (int*)  (ws + 8192);                    // 256 KB
    int*   counts  = (int*)  (ws + 8192 + NROWS * 4);        //  8 KB
    float* partial = (float*)(ws + 8192 + NROWS * 4 + 8192); // 32 KB (8192 blocks)

    vq_seed_kernel    <<<NCODE / 256, 256, 0, stream>>>(cb, seed);
    vq_init_kernel    <<<NCODE / 256, 256, 0, stream>>>(counts);
    vq_argmax_kernel  <<<NROWS / 128, 256, 0, stream>>>(x, cb, seed, idx);
    vq_quantize_kernel<<<NROWS / 8,   256, 0, stream>>>(x, cb, idx, out, counts, partial);
    vq_finalize_kernel<<<1,           256, 0, stream>>>(counts, partial,
                                                        out + (long)NROWS * DIM,
                                                        out + (long)NROWS * DIM + 1);
}